// GraphSAGENet_67353677136305
// MI455X (gfx1250) — compile-verified
//
#include <hip/hip_runtime.h>

#define N_NODES 100000
#define N_EDGES 1200000
#define IN_CH   64
#define HID     16
#define OUT_CH  64

typedef float v2f __attribute__((ext_vector_type(2)));
typedef float v8f __attribute__((ext_vector_type(8)));

__device__ __forceinline__ int clampi(int v, int lo, int hi) {
    return v < lo ? lo : (v > hi ? hi : v);
}

// ---------------------------------------------------------------------------
// Zero a float buffer.
__global__ void zero_f32(float* __restrict__ p, size_t n) {
    size_t i = (size_t)blockIdx.x * blockDim.x + threadIdx.x;
    if (i < n) p[i] = 0.0f;
}

// ---------------------------------------------------------------------------
// Conv1 edge pass: msg = x[src] + edge_attr @ W1e + b1e ; scatter-add to agg1.
// One wave per edge; lane handles channels {2*lane, 2*lane+1} (b64 loads).
// Edge id forced wave-uniform so index/attr loads scalarize to SMEM.
__global__ void edge_pass1(const float* __restrict__ x,
                           const int*   __restrict__ ei,   // [2, E]
                           const float* __restrict__ ea,   // [E, 3]
                           const float* __restrict__ W1e,  // [3, 64]
                           const float* __restrict__ b1e,  // [64]
                           float* __restrict__ agg1,       // [N, 64]
                           float* __restrict__ deg) {      // [N]
    int wave = (int)((blockIdx.x * (size_t)blockDim.x + threadIdx.x) >> 5);
    int lane = threadIdx.x & 31;
    if (wave >= N_EDGES) return;                 // wave-uniform
    int e = __builtin_amdgcn_readfirstlane(wave);

    int src = clampi(ei[e],           0, N_NODES - 1);
    int dst = clampi(ei[N_EDGES + e], 0, N_NODES - 1);
    float e0 = ea[(size_t)e * 3 + 0];
    float e1 = ea[(size_t)e * 3 + 1];
    float e2 = ea[(size_t)e * 3 + 2];
    if (lane == 0) atomicAdd(&deg[dst], 1.0f);

    int c = lane * 2;
    float2 xv = ((const float2*)(x + (size_t)src * IN_CH))[lane];
    float2 w0 = ((const float2*)(W1e      ))[lane];
    float2 w1 = ((const float2*)(W1e +  64))[lane];
    float2 w2 = ((const float2*)(W1e + 128))[lane];
    float2 bb = ((const float2*)(b1e      ))[lane];
    float m0 = xv.x + e0 * w0.x + e1 * w1.x + e2 * w2.x + bb.x;
    float m1 = xv.y + e0 * w0.y + e1 * w1.y + e2 * w2.y + bb.y;
    atomicAdd(&agg1[(size_t)dst * IN_CH + c],     m0);
    atomicAdd(&agg1[(size_t)dst * IN_CH + c + 1], m1);
}

// ---------------------------------------------------------------------------
// Conv1 node GEMM: h = relu((agg1/deg) @ W1l + x @ W1r + b1l + b1r)
// One wave computes one 16-node x 16-hidden tile via V_WMMA_F32_16X16X4_F32.
// Weights staged in LDS once per block (natural stride 16 is conflict-free).
__global__ void __launch_bounds__(256)
node_gemm1(const float* __restrict__ x,     // [N, 64]
           const float* __restrict__ agg1,  // [N, 64]
           const float* __restrict__ deg,   // [N]
           const float* __restrict__ W1l,   // [64, 16]
           const float* __restrict__ b1l,   // [16]
           const float* __restrict__ W1r,   // [64, 16]
           const float* __restrict__ b1r,   // [16]
           float* __restrict__ h) {         // [N, 16]
    __shared__ float sL[IN_CH * HID];       // 1024 floats
    __shared__ float sR[IN_CH * HID];

    const int ntiles = N_NODES / 16;        // 6250 (exact)
    int tid  = threadIdx.x;
    int wave = (int)((blockIdx.x * (size_t)blockDim.x + tid) >> 5);
    int lane = tid & 31;

    // Stage weights (ALL threads participate before any early-out).
#pragma unroll
    for (int i = tid; i < IN_CH * HID; i += 256) {
        sL[i] = W1l[i];
        sR[i] = W1r[i];
    }
    __syncthreads();
    if (wave >= ntiles) return;             // wave-uniform -> EXEC all 1s below

    int rowA  = lane & 15;                  // A: M index ; also B: N column
    int khalf = (lane >> 4) * 2;            // A/B: K sub-offset (0 or 2)
    size_t nodeA = (size_t)wave * 16 + rowA;
    float invd = __builtin_amdgcn_rcpf(fmaxf(deg[nodeA], 1.0f));

    v8f acc = {};
    // (agg1 * invd) @ W1l : K = 64 in 16 steps of 4
#pragma unroll
    for (int ks = 0; ks < 16; ++ks) {
        int k0 = ks * 4 + khalf;
        v2f a; a.x = agg1[nodeA * IN_CH + k0] * invd;
               a.y = agg1[nodeA * IN_CH + k0 + 1] * invd;
        v2f b; b.x = sL[k0 * HID + rowA];
               b.y = sL[(k0 + 1) * HID + rowA];
        acc = __builtin_amdgcn_wmma_f32_16x16x4_f32(false, a, false, b,
                                                    (short)0, acc, false, false);
    }
    // x @ W1r
#pragma unroll
    for (int ks = 0; ks < 16; ++ks) {
        int k0 = ks * 4 + khalf;
        v2f a; a.x = x[nodeA * IN_CH + k0];
               a.y = x[nodeA * IN_CH + k0 + 1];
        v2f b; b.x = sR[k0 * HID + rowA];
               b.y = sR[(k0 + 1) * HID + rowA];
        acc = __builtin_amdgcn_wmma_f32_16x16x4_f32(false, a, false, b,
                                                    (short)0, acc, false, false);
    }
    // D layout: vgpr v, lane l -> M = v + (l>>4)*8, N = l&15
    int colD    = lane & 15;
    int rowBase = (lane >> 4) * 8;
    float bias  = b1l[colD] + b1r[colD];
#pragma unroll
    for (int v = 0; v < 8; ++v) {
        size_t node = (size_t)wave * 16 + rowBase + v;
        h[node * HID + colD] = fmaxf(acc[v] + bias, 0.0f);
    }
}

// ---------------------------------------------------------------------------
// Conv2 edge pass: msg2 = h[src] + edge_attr @ W2e + b2e ; scatter-add to agg2.
// One thread per (edge, channel).
__global__ void edge_pass2(const float* __restrict__ hbuf,  // [N, 16]
                           const int*   __restrict__ ei,
                           const float* __restrict__ ea,
                           const float* __restrict__ W2e,   // [3, 16]
                           const float* __restrict__ b2e,   // [16]
                           float* __restrict__ agg2) {      // [N, 16]
    size_t t = (size_t)blockIdx.x * blockDim.x + threadIdx.x;
    if (t >= (size_t)N_EDGES * HID) return;
    int e = (int)(t >> 4);
    int c = (int)(t & 15);
    int src = clampi(ei[e],           0, N_NODES - 1);
    int dst = clampi(ei[N_EDGES + e], 0, N_NODES - 1);
    float m = hbuf[(size_t)src * HID + c]
            + ea[(size_t)e * 3 + 0] * W2e[c]
            + ea[(size_t)e * 3 + 1] * W2e[16 + c]
            + ea[(size_t)e * 3 + 2] * W2e[32 + c]
            + b2e[c];
    atomicAdd(&agg2[(size_t)dst * HID + c], m);
}

// ---------------------------------------------------------------------------
// Conv2 node GEMM: out = (agg2/deg) @ W2l + h @ W2r + b2l + b2r
// One wave per 16-node tile; 4 output col-tiles of 16, K = 16 in 4 steps.
// Weights staged in LDS with row stride 80 (=64+16) so the two half-wave
// K-offsets (delta = 2 rows = 160 words = 32 banks) never collide.
#define W2_STRIDE 80
__global__ void __launch_bounds__(256)
node_gemm2(const float* __restrict__ hbuf,  // [N, 16]
           const float* __restrict__ agg2,  // [N, 16]
           const float* __restrict__ deg,   // [N]
           const float* __restrict__ W2l,   // [16, 64]
           const float* __restrict__ b2l,   // [64]
           const float* __restrict__ W2r,   // [16, 64]
           const float* __restrict__ b2r,   // [64]
           float* __restrict__ out) {       // [N, 64]
    __shared__ float sL[HID * W2_STRIDE];   // 1280 floats
    __shared__ float sR[HID * W2_STRIDE];

    const int ntiles = N_NODES / 16;
    int tid  = threadIdx.x;
    int wave = (int)((blockIdx.x * (size_t)blockDim.x + tid) >> 5);
    int lane = tid & 31;

#pragma unroll
    for (int i = tid; i < HID * OUT_CH; i += 256) {
        int k = i >> 6, c = i & 63;
        sL[k * W2_STRIDE + c] = W2l[i];
        sR[k * W2_STRIDE + c] = W2r[i];
    }
    __syncthreads();
    if (wave >= ntiles) return;             // wave-uniform

    int rowA  = lane & 15;
    int khalf = (lane >> 4) * 2;
    size_t nodeA = (size_t)wave * 16 + rowA;
    float invd = __builtin_amdgcn_rcpf(fmaxf(deg[nodeA], 1.0f));

    // Preload A fragments for all 4 K-steps (shared across the 4 col-tiles).
    v2f aAgg[4], aH[4];
#pragma unroll
    for (int ks = 0; ks < 4; ++ks) {
        int k0 = ks * 4 + khalf;
        aAgg[ks].x = agg2[nodeA * HID + k0] * invd;
        aAgg[ks].y = agg2[nodeA * HID + k0 + 1] * invd;
        aH[ks].x   = hbuf[nodeA * HID + k0];
        aH[ks].y   = hbuf[nodeA * HID + k0 + 1];
    }

    int colN    = lane & 15;
    int rowBase = (lane >> 4) * 8;
#pragma unroll
    for (int j = 0; j < 4; ++j) {           // output column tile (16 cols each)
        v8f acc = {};
#pragma unroll
        for (int ks = 0; ks < 4; ++ks) {
            int k0 = ks * 4 + khalf;
            v2f bl; bl.x = sL[k0 * W2_STRIDE + j * 16 + colN];
                    bl.y = sL[(k0 + 1) * W2_STRIDE + j * 16 + colN];
            acc = __builtin_amdgcn_wmma_f32_16x16x4_f32(false, aAgg[ks], false, bl,
                                                        (short)0, acc, false, false);
            v2f br; br.x = sR[k0 * W2_STRIDE + j * 16 + colN];
                    br.y = sR[(k0 + 1) * W2_STRIDE + j * 16 + colN];
            acc = __builtin_amdgcn_wmma_f32_16x16x4_f32(false, aH[ks], false, br,
                                                        (short)0, acc, false, false);
        }
        int colD   = j * 16 + colN;
        float bias = b2l[colD] + b2r[colD];
#pragma unroll
        for (int v = 0; v < 8; ++v) {
            size_t node = (size_t)wave * 16 + rowBase + v;
            out[node * OUT_CH + colD] = acc[v] + bias;
        }
    }
}

// ---------------------------------------------------------------------------
extern "C" void kernel_launch(void* const* d_in, const int* in_sizes, int n_in,
                              void* d_out, int out_size, void* d_ws, size_t ws_size,
                              hipStream_t stream) {
    (void)in_sizes; (void)n_in; (void)out_size; (void)ws_size;

    const float* x   = (const float*)d_in[0];
    const int*   ei  = (const int*)  d_in[1];   // edge_index [2, E]
    const float* ea  = (const float*)d_in[2];   // edge_attr  [E, 3]
    const float* W1l = (const float*)d_in[3];
    const float* b1l = (const float*)d_in[4];
    const float* W1r = (const float*)d_in[5];
    const float* b1r = (const float*)d_in[6];
    const float* W1e = (const float*)d_in[7];
    const float* b1e = (const float*)d_in[8];
    const float* W2l = (const float*)d_in[9];
    const float* b2l = (const float*)d_in[10];
    const float* W2r = (const float*)d_in[11];
    const float* b2r = (const float*)d_in[12];
    const float* W2e = (const float*)d_in[13];
    const float* b2e = (const float*)d_in[14];
    float* out = (float*)d_out;

    // Workspace layout (floats): agg1[N*64] | deg[N] | agg2[N*16] | h[N*16]
    float* ws   = (float*)d_ws;
    float* agg1 = ws;
    float* deg  = agg1 + (size_t)N_NODES * IN_CH;
    float* agg2 = deg  + (size_t)N_NODES;
    float* hbuf = agg2 + (size_t)N_NODES * HID;

    const int BLK = 256;

    // 0) zero accumulators (agg1, deg, agg2 are contiguous)
    size_t nz = (size_t)N_NODES * (IN_CH + 1 + HID);
    zero_f32<<<(unsigned)((nz + BLK - 1) / BLK), BLK, 0, stream>>>(agg1, nz);

    // 1) conv1 edge pass: one wave per edge
    {
        size_t nthreads = (size_t)N_EDGES * 32;
        edge_pass1<<<(unsigned)((nthreads + BLK - 1) / BLK), BLK, 0, stream>>>(
            x, ei, ea, W1e, b1e, agg1, deg);
    }

    // 2) conv1 node GEMM (WMMA): h = relu(mean_agg @ W1l + x @ W1r + biases)
    {
        size_t nthreads = (size_t)(N_NODES / 16) * 32;
        node_gemm1<<<(unsigned)((nthreads + BLK - 1) / BLK), BLK, 0, stream>>>(
            x, agg1, deg, W1l, b1l, W1r, b1r, hbuf);
    }

    // 3) conv2 edge pass: one thread per (edge, channel)
    {
        size_t nthreads = (size_t)N_EDGES * HID;
        edge_pass2<<<(unsigned)((nthreads + BLK - 1) / BLK), BLK, 0, stream>>>(
            hbuf, ei, ea, W2e, b2e, agg2);
    }

    // 4) conv2 node GEMM (WMMA): out = mean_agg2 @ W2l + h @ W2r + biases
    {
        size_t nthreads = (size_t)(N_NODES / 16) * 32;
        node_gemm2<<<(unsigned)((nthreads + BLK - 1) / BLK), BLK, 0, stream>>>(
            hbuf, agg2, deg, W2l, b2l, W2r, b2r, out);
    }
}